// LSEP_65678639890840
// MI455X (gfx1250) — compile-verified
//
#include <hip/hip_runtime.h>
#include <math.h>

// ---------------------------------------------------------------------------
// LSEP loss, factored form:
//   per row: s_pos = sum_{y==1} exp(-x), s_neg = sum_{y!=1} exp(x)
//   out = mean_n log(1 + s_pos*s_neg)
//
// MI455X reasoning: 8.4 MB compulsory traffic @ 23.3 TB/s ~= 0.36us -> this
// is a bandwidth-bound two-level reduction. Key points:
//  * ONE v_exp_f32 per element: select the sign of x before the exp
//    (exp(-x) for positives, exp(x) for negatives), instead of computing
//    both arms and masking -> halves the dominant transcendental work.
//  * f32 accumulation throughout (exp sums need f32), so the CDNA5 matrix
//    path is used where it genuinely fits: V_WMMA_F32_16X16X4_F32 with
//    B = ones computes all 16 row sums of the wave's 16x4 f32 partial
//    matrix in one op (lane-local accumulators are laid out exactly as the
//    ISA's 16x4 f32 A-matrix: lanes 0-15 hold K=0,1; lanes 16-31 hold K=2,3).
//  * Streaming data is read exactly once -> non-temporal loads (TH=NT) keep
//    it from polluting WGP$/L2.
//  * 256 waves spread as 64 blocks x 128 threads (1 wave/SIMD over 64 WGPs).
// ---------------------------------------------------------------------------

typedef __attribute__((ext_vector_type(2))) float v2f;
typedef __attribute__((ext_vector_type(8))) float v8f;
typedef __attribute__((ext_vector_type(4))) float f32x4;
typedef __attribute__((ext_vector_type(4))) int   i32x4;

#define LSEP_N 4096
#define LSEP_C 256
#define WAVES_TOTAL (LSEP_N / 16)   // 256 waves, 16 rows per wave

__global__ __launch_bounds__(128) void lsep_rows_kernel(
    const float* __restrict__ y_hat,
    const int*   __restrict__ y,
    float*       __restrict__ partial)   // [WAVES_TOTAL]
{
    const int lane        = threadIdx.x & 31;
    const int waveInBlock = threadIdx.x >> 5;
    const int wave        = blockIdx.x * 4 + waveInBlock;   // 0..255
    const int rowBase     = wave * 16;

    // A-matrix slot for this lane: row = rowBase + (lane&15),
    // K-pair = {0,1} for lanes 0-15 (cols 0..127), {2,3} for lanes 16-31 (cols 128..255)
    const int row     = rowBase + (lane & 15);
    const int khalf   = lane >> 4;
    const int colBase = khalf * 128;

    const float* rowYh = y_hat + (size_t)row * LSEP_C + colBase;
    const int*   rowY  = y     + (size_t)row * LSEP_C + colBase;

    float accp0 = 0.f, accp1 = 0.f;   // pos stream, K slots 0/1 of this lane
    float accn0 = 0.f, accn1 = 0.f;   // neg stream

    #pragma unroll
    for (int i = 0; i < 16; ++i) {
        f32x4 x0 = __builtin_nontemporal_load((const f32x4*)(rowYh + i * 4));
        i32x4 m0 = __builtin_nontemporal_load((const i32x4*)(rowY  + i * 4));
        f32x4 x1 = __builtin_nontemporal_load((const f32x4*)(rowYh + 64 + i * 4));
        i32x4 m1 = __builtin_nontemporal_load((const i32x4*)(rowY  + 64 + i * 4));
        #pragma unroll
        for (int e = 0; e < 4; ++e) {
            {   // cols [colBase .. colBase+63]
                const bool  pa = (m0[e] == 1);
                const float ea = __expf(pa ? -x0[e] : x0[e]);  // single exp/elem
                accp0 += pa ? ea : 0.0f;
                accn0 += pa ? 0.0f : ea;
            }
            {   // cols [colBase+64 .. colBase+127]
                const bool  pb = (m1[e] == 1);
                const float eb = __expf(pb ? -x1[e] : x1[e]);
                accp1 += pb ? eb : 0.0f;
                accn1 += pb ? 0.0f : eb;
            }
        }
    }

    // 16x4 f32 A-matrix, B = ones(4x16)  ->  D[m][n] = sum_k A[m][k] = row sum.
    v2f aPos = {accp0, accp1};
    v2f aNeg = {accn0, accn1};
    v2f bOne = {1.0f, 1.0f};
    v8f zero = {};

    v8f dPos = __builtin_amdgcn_wmma_f32_16x16x4_f32(
        /*neg_a=*/false, aPos, /*neg_b=*/false, bOne,
        /*c_mod=*/(short)0, zero, /*reuse_a=*/false, /*reuse_b=*/false);
    v8f dNeg = __builtin_amdgcn_wmma_f32_16x16x4_f32(
        /*neg_a=*/false, aNeg, /*neg_b=*/false, bOne,
        /*c_mod=*/(short)0, zero, /*reuse_a=*/false, /*reuse_b=*/false);

    // D layout: VGPR v, lanes 0-15 -> row rowBase+v; lanes 16-31 -> row rowBase+v+8.
    float s = 0.0f;
    #pragma unroll
    for (int v = 0; v < 8; ++v)
        s += __logf(1.0f + dPos[v] * dNeg[v]);

    // lanes <16 hold sum over rows 0..7 of the tile; lanes >=16 hold rows 8..15.
    float tileSum = s + __shfl_xor(s, 16, 32);
    if (lane == 0) partial[wave] = tileSum;   // every slot overwritten each call
}

__global__ __launch_bounds__(32) void lsep_final_kernel(
    const float* __restrict__ partial,
    float*       __restrict__ out)
{
    const int lane = threadIdx.x;   // one wave of 32
    float s = 0.0f;
    #pragma unroll
    for (int i = 0; i < WAVES_TOTAL / 32; ++i)
        s += partial[lane + i * 32];
    #pragma unroll
    for (int off = 16; off >= 1; off >>= 1)
        s += __shfl_xor(s, off, 32);
    if (lane == 0) out[0] = s * (1.0f / (float)LSEP_N);
}

extern "C" void kernel_launch(void* const* d_in, const int* in_sizes, int n_in,
                              void* d_out, int out_size, void* d_ws, size_t ws_size,
                              hipStream_t stream) {
    const float* y_hat = (const float*)d_in[0];   // [4096, 256] f32
    const int*   y     = (const int*)d_in[1];     // [4096, 256] i32
    float* out     = (float*)d_out;               // scalar f32
    float* partial = (float*)d_ws;                // 256 floats of scratch

    // 256 waves total: 64 blocks x 128 threads (4 waves/block -> 1 wave/SIMD
    // across 64 WGPs)
    lsep_rows_kernel<<<WAVES_TOTAL / 4, 128, 0, stream>>>(y_hat, y, partial);
    lsep_final_kernel<<<1, 32, 0, stream>>>(partial, out);
}